// BathtubReconstructor_44530220924998
// MI455X (gfx1250) — compile-verified
//
#include <hip/hip_runtime.h>
#include <stdint.h>

#define MAX_ITERS 20

// One block per coarse (y,x) cell: 128 threads = 4 wave32s.
// Each thread owns one b and FOUR consecutive t values (tg = t/4):
//   lane = { b_low(2) , tg(3) }  -> 128B contiguous loads/stores per b, 4 runs per wave
//   b    = wave*4 + (lane>>3)
// The 4 bisections per thread share one set of 16 z registers and give the
// scheduler 4 independent chains for VOPD dual-issue.
// Each wave stages the 4x4 topo tile (64B) into its private LDS slot with
// global_load_async_to_lds_b128 (CDNA5 async path, ASYNCcnt).
__global__ __launch_bounds__(128)
void bathtub_reconstruct_kernel(const float* __restrict__ u_coarse,
                                const float* __restrict__ topo,
                                float* __restrict__ out) {
    __shared__ __align__(16) float zsh[4 * 16];    // 4 waves x 16 z values

    const unsigned tid  = threadIdx.x;
    const unsigned lane = tid & 31u;
    const unsigned wave = tid >> 5;                // 0..3
    const unsigned tg   = lane & 7u;               // t = 4*tg .. 4*tg+3
    const unsigned b    = (wave << 2) | (lane >> 3);
    const unsigned y    = blockIdx.x >> 6;         // 0..63
    const unsigned x    = blockIdx.x & 63u;        // 0..63

    // ---- async global->LDS copy of this cell's 4x4 topo tile -------------
    if (lane < 4u) {
        unsigned goff  = ((y * 4u + lane) * 256u + x * 4u) * 4u;   // byte offset
        unsigned laddr = (unsigned)(uintptr_t)(&zsh[wave * 16u + lane * 4u]);
        asm volatile("global_load_async_to_lds_b128 %0, %1, %2"
                     :: "v"(laddr), "v"(goff), "s"(topo)
                     : "memory");
    }

    // overlap: load the 4 d_targets (one b128) while the copy is in flight
    const float4 dtv =
        *(const float4*)&u_coarse[((b * 64u + y) * 64u + x) * 32u + tg * 4u];
    const float dt[4] = {dtv.x, dtv.y, dtv.z, dtv.w};

    asm volatile("s_wait_asynccnt 0" ::: "memory");

    // ---- pull the 16 z values into VGPRs (broadcast ds_load_b128 x4) -----
    const float4* zv = (const float4*)&zsh[wave * 16u];
    float zr[16];
    {
        float4 q0 = zv[0], q1 = zv[1], q2 = zv[2], q3 = zv[3];
        zr[0]=q0.x; zr[1]=q0.y; zr[2]=q0.z; zr[3]=q0.w;
        zr[4]=q1.x; zr[5]=q1.y; zr[6]=q1.z; zr[7]=q1.w;
        zr[8]=q2.x; zr[9]=q2.y; zr[10]=q2.z; zr[11]=q2.w;
        zr[12]=q3.x; zr[13]=q3.y; zr[14]=q3.z; zr[15]=q3.w;
    }

    // per-cell scalars: zmin, zmax, Zs = sum(z)
    float zmin = zr[0], zmax = zr[0], Zs = zr[0];
#pragma unroll
    for (int k = 1; k < 16; ++k) {
        zmin = fminf(zmin, zr[k]);
        zmax = fmaxf(zmax, zr[k]);
        Zs  += zr[k];
    }

    // Bisection compare:  mean(relu(h-z)) < dt  <=>  Sum|h-z_k| < 32*dt + Zs - 16*h
    float K[4], hl[4], hh[4];
#pragma unroll
    for (int c = 0; c < 4; ++c) {
        K[c]  = fmaf(32.0f, dt[c], Zs);
        hl[c] = zmin;
        hh[c] = (zmax + dt[c]) + 0.001f;
    }

#pragma unroll
    for (int it = 0; it < MAX_ITERS; ++it) {
#pragma unroll
        for (int c = 0; c < 4; ++c) {
            const float h = 0.5f * (hl[c] + hh[c]);
            float s0 = __builtin_fabsf(h - zr[0]);
            float s1 = __builtin_fabsf(h - zr[1]);
            float s2 = __builtin_fabsf(h - zr[2]);
            float s3 = __builtin_fabsf(h - zr[3]);
#pragma unroll
            for (int k = 4; k < 16; k += 4) {
                s0 += __builtin_fabsf(h - zr[k + 0]);
                s1 += __builtin_fabsf(h - zr[k + 1]);
                s2 += __builtin_fabsf(h - zr[k + 2]);
                s3 += __builtin_fabsf(h - zr[k + 3]);
            }
            const float S   = (s0 + s1) + (s2 + s3);
            const float rhs = fmaf(-16.0f, h, K[c]);
            const bool  m   = S < rhs;          // d_mid < d_target
            hl[c] = m ? h     : hl[c];
            hh[c] = m ? hh[c] : h;
        }
    }

    float h[4];
#pragma unroll
    for (int c = 0; c < 4; ++c) h[c] = 0.5f * (hl[c] + hh[c]);

    // ---- scatter fine output: out[b][4y+fy][4x+fx][4*tg .. +3] -----------
    const unsigned obase = ((b * 256u + y * 4u) * 256u + x * 4u) * 32u + tg * 4u;
#pragma unroll
    for (int fy = 0; fy < 4; ++fy) {
#pragma unroll
        for (int fx = 0; fx < 4; ++fx) {
            const float zk = zr[fy * 4 + fx];
            float4 v;
            v.x = fmaxf(h[0] - zk, 0.0f);
            v.y = fmaxf(h[1] - zk, 0.0f);
            v.z = fmaxf(h[2] - zk, 0.0f);
            v.w = fmaxf(h[3] - zk, 0.0f);
            *(float4*)&out[obase + (unsigned)(fy * 256 + fx) * 32u] = v;
        }
    }
}

extern "C" void kernel_launch(void* const* d_in, const int* in_sizes, int n_in,
                              void* d_out, int out_size, void* d_ws, size_t ws_size,
                              hipStream_t stream) {
    const float* u_coarse = (const float*)d_in[0];   // (16, 64, 64, 32) fp32
    const float* topo     = (const float*)d_in[1];   // (256, 256) fp32
    float*       out      = (float*)d_out;           // (16, 256, 256, 32) fp32
    (void)in_sizes; (void)n_in; (void)out_size; (void)d_ws; (void)ws_size;

    dim3 grid(64 * 64);   // one block per coarse (y,x)
    dim3 block(128);      // 4 wave32s: thread = (b, 4 t's)
    bathtub_reconstruct_kernel<<<grid, block, 0, stream>>>(u_coarse, topo, out);
}